// LDAMMarginLoss_19301583028311
// MI455X (gfx1250) — compile-verified
//
#include <hip/hip_runtime.h>
#include <math.h>

typedef __attribute__((ext_vector_type(2))) float v2f;
typedef __attribute__((ext_vector_type(4))) float v4f;
typedef __attribute__((ext_vector_type(4))) int   v4i;
typedef __attribute__((ext_vector_type(8))) float v8f;

// Fixed block shape: 256 threads = 8 wave32 waves.
#define KTHREADS 256
#define KWAVES   8

// ---------------------------------------------------------------------------
// Wave32 all-lane sum using the CDNA5 matrix unit.
// A (16x4): lane m holds A[m][0]=p, A[m][1]=0 ; lane m+16 holds A[m][2]=p, A[m][3]=0
// B (4x16): all ones.
// D[m][n] = p_m + p_{m+16}  (independent of n)
// Lane L in [0,15]:  d[j] = row j   -> sum_j d[j] = rows 0..7
// Lane L in [16,31]: d[j] = row j+8 -> sum_j d[j] = rows 8..15
// One xor-16 shuffle-add yields the full 32-lane sum in every lane.
// EXEC must be all ones at the call site (full wave, no divergence).
// ---------------------------------------------------------------------------
__device__ __forceinline__ float wave_sum32(float p) {
    v2f a; a.x = p;    a.y = 0.0f;
    v2f b; b.x = 1.0f; b.y = 1.0f;
    v8f c = {};
    v8f d = __builtin_amdgcn_wmma_f32_16x16x4_f32(
        /*neg_a=*/false, a, /*neg_b=*/false, b,
        /*c_mod=*/(short)0, c, /*reuse_a=*/false, /*reuse_b=*/false);
    float s = d[0] + d[1] + d[2] + d[3] + d[4] + d[5] + d[6] + d[7];
    s += __shfl_xor(s, 16, 32);
    return s;
}

__device__ __forceinline__ float loss_elem(float x, int t, float m0, float m1) {
    // adj = logit + (t==1 ? -m1 : +m0)
    float adj = x + ((t == 1) ? -m1 : m0);
    // stable BCE-with-logits: max(adj,0) - adj*t + log1p(exp(-|adj|))
    return fmaxf(adj, 0.0f) - adj * (float)t + log1pf(expf(-fabsf(adj)));
}

__device__ __forceinline__ float loss4(const v4f& x, const v4i& t, float m0, float m1) {
    float a = loss_elem(x.x, t.x, m0, m1);
    float b = loss_elem(x.y, t.y, m0, m1);
    float c = loss_elem(x.z, t.z, m0, m1);
    float d = loss_elem(x.w, t.w, m0, m1);
    return (a + b) + (c + d);
}

// Block reduction: wave32 WMMA tree, then fixed 8-wave LDS combine (thread 0).
__device__ __forceinline__ void block_store_sum(float acc, float* dst, float scale) {
    __shared__ float lds[KWAVES];
    float w = wave_sum32(acc);
    const unsigned lane = threadIdx.x & 31u;
    const unsigned wid  = threadIdx.x >> 5;
    if (lane == 0) lds[wid] = w;
    __syncthreads();
    if (threadIdx.x == 0) {
        float s = ((lds[0] + lds[1]) + (lds[2] + lds[3])) +
                  ((lds[4] + lds[5]) + (lds[6] + lds[7]));
        *dst = s * scale;
    }
}

// ---------------------------------------------------------------------------
// Stage 1: grid-stride streaming pass, NT 128-bit loads, per-block partial sum.
// 32-bit indexing (B = 2^25 << 2^31).
// ---------------------------------------------------------------------------
__global__ void __launch_bounds__(KTHREADS)
ldam_partial_kernel(const float* __restrict__ logits,
                    const int*   __restrict__ targets,
                    const float* __restrict__ m_list,
                    float*       __restrict__ partials,
                    unsigned int n, unsigned int nvec) {
    const float m0 = m_list[0];
    const float m1 = m_list[1];

    const unsigned int gid    = blockIdx.x * (unsigned)blockDim.x + threadIdx.x;
    const unsigned int stride = gridDim.x * (unsigned)blockDim.x;

    float acc = 0.0f;
    const v4f* __restrict__ lv = (const v4f*)logits;
    const v4i* __restrict__ tv = (const v4i*)targets;

    // Unrolled x2: four 128-bit NT loads in flight before any math.
    unsigned int i = gid;
    for (; i + stride < nvec; i += 2u * stride) {
        v4f x0 = __builtin_nontemporal_load(lv + i);            // global_load_b128 NT
        v4i t0 = __builtin_nontemporal_load(tv + i);
        v4f x1 = __builtin_nontemporal_load(lv + i + stride);
        v4i t1 = __builtin_nontemporal_load(tv + i + stride);
        acc += loss4(x0, t0, m0, m1);
        acc += loss4(x1, t1, m0, m1);
    }
    if (i < nvec) {
        v4f x = __builtin_nontemporal_load(lv + i);
        v4i t = __builtin_nontemporal_load(tv + i);
        acc += loss4(x, t, m0, m1);
    }
    // scalar tail (empty when n % 4 == 0)
    for (unsigned int j = 4u * nvec + gid; j < n; j += stride)
        acc += loss_elem(logits[j], targets[j], m0, m1);

    block_store_sum(acc, &partials[blockIdx.x], 1.0f);
}

// ---------------------------------------------------------------------------
// Stage 2: single block folds the per-block partials and writes the mean.
// ---------------------------------------------------------------------------
__global__ void __launch_bounds__(KTHREADS)
ldam_final_kernel(const float* __restrict__ partials, unsigned int np,
                  float* __restrict__ out, float inv_n) {
    float acc = 0.0f;
    for (unsigned int i = threadIdx.x; i < np; i += KTHREADS)
        acc += partials[i];
    block_store_sum(acc, out, inv_n);
}

extern "C" void kernel_launch(void* const* d_in, const int* in_sizes, int n_in,
                              void* d_out, int out_size, void* d_ws, size_t ws_size,
                              hipStream_t stream) {
    // setup_inputs order: logits (f32[B]), m_list (f32[2]), targets (i32[B])
    const float* logits  = (const float*)d_in[0];
    const float* m_list  = (const float*)d_in[1];
    const int*   targets = (const int*)d_in[2];

    const unsigned int n    = (unsigned int)in_sizes[0];
    const unsigned int nvec = n >> 2;

    int nblocks = 2048;                  // ~16 float4 iters/thread at B = 2^25
    unsigned int maxb = (nvec + KTHREADS - 1) / KTHREADS;
    if (maxb < 1u) maxb = 1u;
    if ((unsigned)nblocks > maxb) nblocks = (int)maxb;
    if ((size_t)nblocks * sizeof(float) > ws_size && ws_size >= sizeof(float))
        nblocks = (int)(ws_size / sizeof(float));
    if (nblocks < 1) nblocks = 1;

    float* partials = (float*)d_ws;

    ldam_partial_kernel<<<nblocks, KTHREADS, 0, stream>>>(
        logits, targets, m_list, partials, n, nvec);
    ldam_final_kernel<<<1, KTHREADS, 0, stream>>>(
        partials, (unsigned int)nblocks, (float*)d_out, 1.0f / (float)n);
}